// GINModel_84473416778175
// MI455X (gfx1250) — compile-verified
//
#include <hip/hip_runtime.h>

#define N_NODESC 20000
#define N_EDGESC 320000
#define HID      64
#define MMC      8
#define CMC      512               // HID * MMC floats per node
#define NFC      (N_NODESC * CMC)  // 10,240,000 floats = 40.96 MB

typedef float v2f __attribute__((ext_vector_type(2)));
typedef float v8f __attribute__((ext_vector_type(8)));

__device__ __forceinline__ v8f wmma_f32(v2f a, v2f b, v8f c) {
    // D = A(16x4) * B(4x16) + C(16x16), all f32, wave32
    return __builtin_amdgcn_wmma_f32_16x16x4_f32(false, a, false, b, (short)0, c, false, false);
}

// ---------------------------------------------------------------- scatter ---
// agg[dst] += h[src], feature = 8 floats (layer 0, C=1)
__global__ __launch_bounds__(256) void scatter8_kernel(
    const float* __restrict__ h, const int* __restrict__ src,
    const int* __restrict__ dst, float* __restrict__ agg)
{
    int i = blockIdx.x * 256 + threadIdx.x;
    if (i >= N_EDGESC * 8) return;
    int e = i >> 3, j = i & 7;
    int s = src[e], d = dst[e];
    atomicAdd(&agg[d * 8 + j], h[s * 8 + j]);
}

// agg[dst] += h[src], feature = 512 floats; thread = (edge, 8-float chunk)
__global__ __launch_bounds__(256) void scatter512_kernel(
    const float* __restrict__ h, const int* __restrict__ src,
    const int* __restrict__ dst, float* __restrict__ agg)
{
    int i = blockIdx.x * 256 + threadIdx.x;
    if (i >= N_EDGESC * 64) return;
    int e = i >> 6, chunk = i & 63;
    int s = src[e], d = dst[e];
    const float4* hp = (const float4*)(h + (size_t)s * CMC + chunk * 8);
    float4 a0 = hp[0], a1 = hp[1];
    float* dp = agg + (size_t)d * CMC + chunk * 8;
    atomicAdd(dp + 0, a0.x); atomicAdd(dp + 1, a0.y);
    atomicAdd(dp + 2, a0.z); atomicAdd(dp + 3, a0.w);
    atomicAdd(dp + 4, a1.x); atomicAdd(dp + 5, a1.y);
    atomicAdd(dp + 6, a1.z); atomicAdd(dp + 7, a1.w);
}

// ------------------------------------------------------------- fused MLP ---
// z = h + agg ; y1 = relu(W1 z + b1) ; y2 = relu(W2 y1 + b2) ; yout = y2
// GEMM over channel axis via V_WMMA_F32_16X16X4_F32.
// Block = 128 threads (4 waves), each wave owns 16 columns (2 nodes x 8 m).
// yout may alias agg (block reads all its agg before any write).
__global__ __launch_bounds__(128) void mlp_kernel(
    const float* __restrict__ h, const float* __restrict__ agg,
    const float* __restrict__ W1, const float* __restrict__ B1,
    const float* __restrict__ W2, const float* __restrict__ B2,
    float* __restrict__ yout, int layer0)
{
    __shared__ float sW1[64 * 65];        // transposed [c][o], pad 65
    __shared__ float sW2[64 * 65];
    __shared__ float sB1[64], sB2[64], sW10[64];
    __shared__ float sY[4][64 * 17 + 4];  // per-wave y1 staging, pad 17

    const int tid = threadIdx.x;
    if (layer0) {
        if (tid < 64) sW10[tid] = W1[tid];
    } else {
        for (int i = tid; i < 4096; i += 128) {
            int o = i >> 6, c = i & 63;
            sW1[c * 65 + o] = W1[i];
        }
    }
    for (int i = tid; i < 4096; i += 128) {
        int o = i >> 6, c = i & 63;
        sW2[c * 65 + o] = W2[i];
    }
    if (tid < 64) { sB1[tid] = B1[tid]; sB2[tid] = B2[tid]; }
    __syncthreads();

    const int wave = tid >> 5;
    const int lane = tid & 31;
    const int half = lane >> 4;   // K offset group (0: K+0, 1: K+2)
    const int l16  = lane & 15;   // column / row within 16
    float* sYw = sY[wave];

    const int gcol = blockIdx.x * 64 + wave * 16 + l16;  // 0..159999
    const int node = gcol >> 3;
    const int m    = gcol & 7;

    const v8f zero8 = {0.f, 0.f, 0.f, 0.f, 0.f, 0.f, 0.f, 0.f};

    if (layer0) {
        float z0 = h[node * 8 + m] + agg[node * 8 + m];
        #pragma unroll
        for (int cc = 0; cc < 32; ++cc) {
            int c = cc + 32 * half;
            sYw[c * 17 + l16] = fmaxf(sW10[c] * z0 + sB1[c], 0.f);
        }
    } else {
        const int base = node * CMC + m;
        float zb0[16], zb1[16];
        #pragma unroll
        for (int g = 0; g < 16; ++g) {
            int c0 = 4 * g + 2 * half;
            zb0[g] = h[base + c0 * 8]       + agg[base + c0 * 8];
            zb1[g] = h[base + (c0 + 1) * 8] + agg[base + (c0 + 1) * 8];
        }
        v8f acc[4] = {zero8, zero8, zero8, zero8};
        #pragma unroll
        for (int g = 0; g < 16; ++g) {
            int c0 = 4 * g + 2 * half;
            v2f b; b[0] = zb0[g]; b[1] = zb1[g];
            #pragma unroll
            for (int t = 0; t < 4; ++t) {
                v2f a;
                a[0] = sW1[c0 * 65 + t * 16 + l16];
                a[1] = sW1[(c0 + 1) * 65 + t * 16 + l16];
                acc[t] = wmma_f32(a, b, acc[t]);
            }
        }
        // bias + relu + stage y1 into LDS (D reg r: rows r / r+8 per half)
        #pragma unroll
        for (int t = 0; t < 4; ++t)
            #pragma unroll
            for (int r = 0; r < 8; ++r) {
                int c = t * 16 + r + 8 * half;
                sYw[c * 17 + l16] = fmaxf(acc[t][r] + sB1[c], 0.f);
            }
    }
    __syncthreads();

    // GEMM2: B from staged y1, A from W2 (transposed in LDS)
    float yb0[16], yb1[16];
    #pragma unroll
    for (int g = 0; g < 16; ++g) {
        int c0 = 4 * g + 2 * half;
        yb0[g] = sYw[c0 * 17 + l16];
        yb1[g] = sYw[(c0 + 1) * 17 + l16];
    }
    v8f acc2[4] = {zero8, zero8, zero8, zero8};
    #pragma unroll
    for (int g = 0; g < 16; ++g) {
        int c0 = 4 * g + 2 * half;
        v2f b; b[0] = yb0[g]; b[1] = yb1[g];
        #pragma unroll
        for (int t = 0; t < 4; ++t) {
            v2f a;
            a[0] = sW2[c0 * 65 + t * 16 + l16];
            a[1] = sW2[(c0 + 1) * 65 + t * 16 + l16];
            acc2[t] = wmma_f32(a, b, acc2[t]);
        }
    }
    const int obase = node * CMC + m;
    #pragma unroll
    for (int t = 0; t < 4; ++t)
        #pragma unroll
        for (int r = 0; r < 8; ++r) {
            int c = t * 16 + r + 8 * half;
            yout[obase + c * 8] = fmaxf(acc2[t][r] + sB2[c], 0.f);
        }
}

// ------------------------------------------------------------- batchnorm ---
__global__ __launch_bounds__(256) void stats_kernel(
    const float* __restrict__ y, float* __restrict__ stats)
{
    __shared__ float sS[64], sQ[64];
    const int tid = threadIdx.x;
    if (tid < 64) { sS[tid] = 0.f; sQ[tid] = 0.f; }
    __syncthreads();
    const float4* y4 = (const float4*)y;
    const int n4 = NFC / 4;
    for (int i = blockIdx.x * 256 + tid; i < n4; i += gridDim.x * 256) {
        float4 v = y4[i];
        int c = (i >> 1) & 63;   // element idx = 4i; channel = (4i/8)%64
        atomicAdd(&sS[c], v.x + v.y + v.z + v.w);
        atomicAdd(&sQ[c], v.x * v.x + v.y * v.y + v.z * v.z + v.w * v.w);
    }
    __syncthreads();
    if (tid < 64) {
        atomicAdd(&stats[tid],      sS[tid]);
        atomicAdd(&stats[64 + tid], sQ[tid]);
    }
}

__global__ void finalize_kernel(const float* __restrict__ stats,
                                const float* __restrict__ gamma,
                                const float* __restrict__ beta,
                                float* __restrict__ ss)
{
    int c = threadIdx.x;
    if (c >= 64) return;
    const float inv = 1.f / (float)(N_NODESC * MMC);
    float mean = stats[c] * inv;
    float var  = stats[64 + c] * inv - mean * mean;
    float rstd = rsqrtf(var + 1e-5f);
    float a = gamma[c] * rstd;
    ss[c]      = a;
    ss[64 + c] = beta[c] - a * mean;
}

__global__ __launch_bounds__(256) void bn_kernel(float* __restrict__ y,
                                                 const float* __restrict__ ss)
{
    int i = blockIdx.x * 256 + threadIdx.x;
    if (i >= NFC / 4) return;
    int c = (i >> 1) & 63;
    float a = ss[c], b = ss[64 + c];
    float4* y4 = (float4*)y;
    float4 v = y4[i];
    v.x = v.x * a + b; v.y = v.y * a + b;
    v.z = v.z * a + b; v.w = v.w * a + b;
    y4[i] = v;
}

// ------------------------------------------------------------------ pool ---
// pooled[c*8+m] = sum_n h[n][c][m] ; 250 blocks x 80 nodes
__global__ __launch_bounds__(256) void pool_kernel(const float* __restrict__ h,
                                                   float* __restrict__ pooled)
{
    const int tid = threadIdx.x;
    const size_t base = (size_t)blockIdx.x * 80 * CMC;
    float a0 = 0.f, a1 = 0.f;
    #pragma unroll 4
    for (int k = 0; k < 160; ++k) {       // 80*512/256
        float v = h[base + k * 256 + tid];
        if (k & 1) a1 += v; else a0 += v; // (k*256+tid)%512 = tid or tid+256
    }
    atomicAdd(&pooled[tid], a0);
    atomicAdd(&pooled[tid + 256], a1);
}

__global__ void out_kernel(const float* __restrict__ pooled,
                           const float* __restrict__ w_out,
                           const float* __restrict__ b_out,
                           float* __restrict__ out)
{
    int m = threadIdx.x;
    if (m >= MMC) return;
    float s = 0.f;
    #pragma unroll
    for (int c = 0; c < 64; ++c) s += w_out[c] * pooled[c * 8 + m];
    s = s / (float)N_NODESC + b_out[0];
    out[m] = 1.f / (1.f + __expf(-s));
}

// ---------------------------------------------------------------- driver ---
extern "C" void kernel_launch(void* const* d_in, const int* in_sizes, int n_in,
                              void* d_out, int out_size, void* d_ws, size_t ws_size,
                              hipStream_t stream)
{
    (void)in_sizes; (void)n_in; (void)out_size; (void)ws_size;
    const float* x     = (const float*)d_in[0];
    const int*   ei    = (const int*)d_in[1];
    const float* w1_0  = (const float*)d_in[3];
    const float* w1    = (const float*)d_in[4];
    const float* b1    = (const float*)d_in[5];
    const float* w2    = (const float*)d_in[6];
    const float* b2    = (const float*)d_in[7];
    const float* gamma = (const float*)d_in[8];
    const float* beta  = (const float*)d_in[9];
    const float* w_out = (const float*)d_in[10];
    const float* b_out = (const float*)d_in[11];
    const int* src = ei;
    const int* dst = ei + N_EDGESC;

    float* wsf    = (float*)d_ws;
    float* bufA   = wsf;                    // 41 MB
    float* bufB   = wsf + NFC;              // 41 MB
    float* agg0   = wsf + 2 * (size_t)NFC;  // 640 KB (layer-0 agg, C=1)
    float* stats  = agg0 + N_NODESC * MMC;  // 128 floats
    float* ss     = stats + 128;            // 128 floats
    float* pooled = ss + 128;               // 512 floats

    const dim3 blk256(256), blk128(128);

    // ---- layer 0 (C=1) ----
    hipMemsetAsync(agg0, 0, (size_t)N_NODESC * MMC * sizeof(float), stream);
    hipMemsetAsync(stats, 0, 128 * sizeof(float), stream);
    scatter8_kernel<<<(N_EDGESC * 8 + 255) / 256, blk256, 0, stream>>>(x, src, dst, agg0);
    mlp_kernel<<<2500, blk128, 0, stream>>>(x, agg0, w1_0, b1, w2, b2, bufA, 1);
    stats_kernel<<<512, blk256, 0, stream>>>(bufA, stats);
    finalize_kernel<<<1, 64, 0, stream>>>(stats, gamma, beta, ss);
    bn_kernel<<<(NFC / 4 + 255) / 256, blk256, 0, stream>>>(bufA, ss);

    // ---- layers 1..4 ----
    float* cur = bufA;
    for (int i = 1; i < 5; ++i) {
        float* nxt = (cur == bufA) ? bufB : bufA;
        hipMemsetAsync(nxt, 0, (size_t)NFC * sizeof(float), stream);
        hipMemsetAsync(stats, 0, 128 * sizeof(float), stream);
        scatter512_kernel<<<(N_EDGESC * 64 + 255) / 256, blk256, 0, stream>>>(cur, src, dst, nxt);
        mlp_kernel<<<2500, blk128, 0, stream>>>(cur, nxt, w1 + (size_t)(i - 1) * 4096,
                                                b1 + i * 64, w2 + (size_t)i * 4096,
                                                b2 + i * 64, nxt, 0);
        stats_kernel<<<512, blk256, 0, stream>>>(nxt, stats);
        finalize_kernel<<<1, 64, 0, stream>>>(stats, gamma + i * 64, beta + i * 64, ss);
        bn_kernel<<<(NFC / 4 + 255) / 256, blk256, 0, stream>>>(nxt, ss);
        cur = nxt;
    }

    // ---- pool + head ----
    hipMemsetAsync(pooled, 0, 512 * sizeof(float), stream);
    pool_kernel<<<250, blk256, 0, stream>>>(cur, pooled);
    out_kernel<<<1, 32, 0, stream>>>(pooled, w_out, b_out, (float*)d_out);
}